// DANClassifier_10015863734695
// MI455X (gfx1250) — compile-verified
//
#include <hip/hip_runtime.h>
#include <hip/hip_bf16.h>
#include <math.h>

// ---------------------------------------------------------------------------
// DAN classifier: gather + segment-mean -> tanh MLP (WMMA f32) -> BCE sum.
// H (hidden dim) is fixed at 128 by the reference model.
// ---------------------------------------------------------------------------

#define HDIM 128

typedef __attribute__((ext_vector_type(2))) float v2f;
typedef __attribute__((ext_vector_type(8))) float v8f;

// ---------------------------------------------------------------------------
// Kernel 1: one wave32 per segment. Binary-search the token range in the
// sorted segment_ids, stream the gathered embedding rows (coalesced 512B per
// token: 32 lanes x float4), mean, write sent[B][128] to workspace.
// ---------------------------------------------------------------------------
__global__ void seg_mean_kernel(const int* __restrict__ token_ids,
                                const int* __restrict__ segment_ids,
                                const float* __restrict__ embed_table,
                                float* __restrict__ sent,
                                int T, int B) {
  const int wave = (int)((blockIdx.x * blockDim.x + threadIdx.x) >> 5);
  const int lane = (int)(threadIdx.x & 31);
  if (wave >= B) return;

  // lower_bound(segment_ids, wave)  (uniform across the wave)
  int lo = 0, hi = T;
  while (lo < hi) { int mid = (lo + hi) >> 1; if (segment_ids[mid] < wave) lo = mid + 1; else hi = mid; }
  const int start = lo;
  // lower_bound(segment_ids, wave + 1)
  hi = T;
  while (lo < hi) { int mid = (lo + hi) >> 1; if (segment_ids[mid] < wave + 1) lo = mid + 1; else hi = mid; }
  const int end = lo;

  const float4* et = (const float4*)embed_table;  // HDIM/4 = 32 float4 per row
  float4 acc = make_float4(0.f, 0.f, 0.f, 0.f);
  for (int t = start; t < end; ++t) {
    const int tok = token_ids[t];
    if (t + 4 < end) {
      // hide dependent-gather latency: prefetch a row 4 tokens ahead
      const int tnext = token_ids[t + 4];
      __builtin_prefetch(&embed_table[(size_t)tnext * HDIM], 0, 3);
    }
    const float4 v = et[(size_t)tok * (HDIM / 4) + lane];
    acc.x += v.x; acc.y += v.y; acc.z += v.z; acc.w += v.w;
  }
  const int cnt = end - start;
  const float scale = (cnt > 0) ? 1.0f / (float)cnt : 0.0f;  // matches sums/max(cnt,1)
  acc.x *= scale; acc.y *= scale; acc.z *= scale; acc.w *= scale;
  ((float4*)sent)[(size_t)wave * (HDIM / 4) + lane] = acc;
}

// ---------------------------------------------------------------------------
// Kernel 2: 16-row slab per block, 8 waves. hid = tanh(sent @ W_hid + b_hid)
// via v_wmma_f32_16x16x4_f32 (K-loop of 32). Epilogue folds in the second
// layer (dot with W_out), sigmoid + clamped-BCE, deterministic per-block sum.
// ---------------------------------------------------------------------------
__global__ void mlp_loss_kernel(const float* __restrict__ sent,
                                const float* __restrict__ W_hid,
                                const float* __restrict__ b_hid,
                                const float* __restrict__ W_out,
                                const float* __restrict__ b_out,
                                const float* __restrict__ y_true,
                                float* __restrict__ block_loss) {
  __shared__ float sA[16 * 132];   // pitch 132 avoids 16-way bank conflicts on A loads
  __shared__ float zp[8][16];      // per-wave second-layer partials

  const int tid  = (int)threadIdx.x;
  const int row0 = (int)blockIdx.x * 16;

  // Stage the 16x128 sent tile into LDS (coalesced).
  for (int i = 0; i < 8; ++i) {
    const int idx = tid + i * 256;
    const int r = idx >> 7, c0 = idx & 127;
    sA[r * 132 + c0] = sent[(size_t)(row0 + r) * HDIM + c0];
  }
  __syncthreads();

  const int lane = tid & 31;
  const int wv   = tid >> 5;        // wave = N-tile index (0..7)
  const int half = lane >> 4;       // lane half selects K pair
  const int mn   = lane & 15;       // A: row M; B: column N within tile
  const int col  = wv * 16 + mn;    // global hidden column (0..127)

  // C/D layout: c[v] = out[M][N], M = v + 8*half, N = col.
  v8f c = {};
  for (int kb = 0; kb < HDIM; kb += 4) {
    const int k0 = kb + 2 * half;   // VGPR v, half h -> K = v + 2h
    v2f a, b;
    a[0] = sA[mn * 132 + k0];
    a[1] = sA[mn * 132 + k0 + 1];
    b[0] = W_hid[(size_t)k0 * HDIM + col];
    b[1] = W_hid[(size_t)(k0 + 1) * HDIM + col];
    c = __builtin_amdgcn_wmma_f32_16x16x4_f32(
        /*neg_a=*/false, a, /*neg_b=*/false, b,
        /*c_mod=*/(short)0, c, /*reuse_a=*/false, /*reuse_b=*/false);
  }

  const float bh = b_hid[col];
  const float wo = W_out[col];
  for (int v = 0; v < 8; ++v) {
    const float hval = tanhf(c[v] + bh);
    float contrib = hval * wo;
    // reduce this wave's 16 N-columns per row (within each 16-lane half)
    for (int off = 1; off < 16; off <<= 1)
      contrib += __shfl_xor(contrib, off, 16);
    if (mn == 0) zp[wv][v + 8 * half] = contrib;
  }
  __syncthreads();

  if (tid < 16) {
    float z = b_out[0];
    for (int w = 0; w < 8; ++w) z += zp[w][tid];     // fixed order: deterministic
    const float p = 1.0f / (1.0f + expf(-z));
    const float y = y_true[row0 + tid];
    const float lp  = fmaxf(logf(p),    -100.0f);
    const float l1p = fmaxf(log1pf(-p), -100.0f);
    float loss = -(y * lp + (1.0f - y) * l1p);
    for (int off = 1; off < 16; off <<= 1)
      loss += __shfl_xor(loss, off, 16);
    if (tid == 0) block_loss[blockIdx.x] = loss;
  }
}

// ---------------------------------------------------------------------------
// Kernel 3: deterministic tree-reduce of per-block losses into d_out[0].
// ---------------------------------------------------------------------------
__global__ void reduce_loss_kernel(const float* __restrict__ block_loss,
                                   float* __restrict__ out, int n) {
  __shared__ float s[256];
  const int tid = (int)threadIdx.x;
  float v = 0.0f;
  for (int i = tid; i < n; i += 256) v += block_loss[i];
  s[tid] = v;
  __syncthreads();
  for (int off = 128; off > 0; off >>= 1) {
    if (tid < off) s[tid] += s[tid + off];
    __syncthreads();
  }
  if (tid == 0) out[0] = s[0];
}

extern "C" void kernel_launch(void* const* d_in, const int* in_sizes, int n_in,
                              void* d_out, int out_size, void* d_ws, size_t ws_size,
                              hipStream_t stream) {
  const int*   token_ids   = (const int*)d_in[0];
  const int*   segment_ids = (const int*)d_in[1];
  const float* y_true      = (const float*)d_in[2];
  const float* embed_table = (const float*)d_in[3];
  const float* W_hid       = (const float*)d_in[4];
  const float* b_hid       = (const float*)d_in[5];
  const float* W_out       = (const float*)d_in[6];
  const float* b_out       = (const float*)d_in[7];

  const int T = in_sizes[0];
  const int B = in_sizes[2];

  float* sent       = (float*)d_ws;              // [B, 128]
  float* block_loss = sent + (size_t)B * HDIM;   // [B/16]

  // Kernel 1: one wave per segment, 8 waves (256 threads) per block.
  const int blk1 = (B + 7) / 8;
  seg_mean_kernel<<<blk1, 256, 0, stream>>>(token_ids, segment_ids, embed_table,
                                            sent, T, B);

  // Kernel 2: one 16-row slab per block.
  const int nblk2 = B / 16;
  mlp_loss_kernel<<<nblk2, 256, 0, stream>>>(sent, W_hid, b_hid, W_out, b_out,
                                             y_true, block_loss);

  // Kernel 3: final scalar.
  reduce_loss_kernel<<<1, 256, 0, stream>>>(block_loss, (float*)d_out, nblk2);
}